// GroupGlobalAttention_3573412790901
// MI455X (gfx1250) — compile-verified
//
#include <hip/hip_runtime.h>
#include <math.h>

typedef float v2f __attribute__((ext_vector_type(2)));
typedef float v8f __attribute__((ext_vector_type(8)));

constexpr int T = 32, B = 16, S = 128, D = 1024;

__device__ __forceinline__ float fast_sigmoid(float x) {
    // sigmoid(x) = 1 / (1 + exp(-x)) ; v_exp_f32 + v_rcp_f32
    return __builtin_amdgcn_rcpf(1.0f + __expf(-x));
}

#define LD2(p) (*(const v2f*)(p))

#define WMMA4(sl)                                                                                  \
    acc0 = __builtin_amdgcn_wmma_f32_16x16x4_f32(false, a##sl, false, s0##sl, (short)0, acc0, false, false); \
    acc1 = __builtin_amdgcn_wmma_f32_16x16x4_f32(false, a##sl, false, s1##sl, (short)0, acc1, false, false); \
    acc2 = __builtin_amdgcn_wmma_f32_16x16x4_f32(false, a##sl, false, s2##sl, (short)0, acc2, false, false); \
    acc3 = __builtin_amdgcn_wmma_f32_16x16x4_f32(false, a##sl, false, s3##sl, (short)0, acc3, false, false)

// ---------------------------------------------------------------------------
// Y[m,n] = sum_k X[m,k] * W[n,k] + bias[n]     (fp32 WMMA 16x16x4, K=D=1024)
// One wave computes a 16x64 output strip; 4-deep software-pipelined loads.
// ---------------------------------------------------------------------------
__global__ __launch_bounds__(256) void proj_gemm_kernel(
    const float* __restrict__ X, const float* __restrict__ W,
    const float* __restrict__ bias, float* __restrict__ Y,
    int M)
{
    constexpr int N = D, K = D;
    const int wid  = (blockIdx.x * blockDim.x + threadIdx.x) >> 5;
    const int lane = threadIdx.x & 31;
    const int half = lane >> 4;      // 0: lanes 0-15, 1: lanes 16-31
    const int r    = lane & 15;
    constexpr int tilesN = N >> 6;
    const int tileM = wid / tilesN;
    const int tileN = wid % tilesN;
    if (tileM * 16 >= M) return;
    const int m0 = tileM * 16;
    const int n0 = tileN * 64;
    const int ko = 2 * half;         // A/B K sub-offset per ISA 16x4 fp32 layout

    const float* Arow = X + (size_t)(m0 + r) * K;
    const float* Br0  = W + (size_t)(n0 +      r) * K;   // B[k][n] = W[n][k]
    const float* Br1  = W + (size_t)(n0 + 16 + r) * K;
    const float* Br2  = W + (size_t)(n0 + 32 + r) * K;
    const float* Br3  = W + (size_t)(n0 + 48 + r) * K;

#define PLOAD(sl, kk)                                                                              \
    a##sl  = LD2(Arow + (kk) + ko);                                                                \
    s0##sl = LD2(Br0  + (kk) + ko);                                                                \
    s1##sl = LD2(Br1  + (kk) + ko);                                                                \
    s2##sl = LD2(Br2  + (kk) + ko);                                                                \
    s3##sl = LD2(Br3  + (kk) + ko)

    v8f acc0 = {}, acc1 = {}, acc2 = {}, acc3 = {};
    v2f a0, s00, s10, s20, s30;
    v2f a1, s01, s11, s21, s31;
    v2f a2, s02, s12, s22, s32;
    v2f a3, s03, s13, s23, s33;

    PLOAD(0, 0);
    PLOAD(1, 4);
    PLOAD(2, 8);
    PLOAD(3, 12);
    for (int k0 = 0; k0 < K - 16; k0 += 16) {
        WMMA4(0); PLOAD(0, k0 + 16);
        WMMA4(1); PLOAD(1, k0 + 20);
        WMMA4(2); PLOAD(2, k0 + 24);
        WMMA4(3); PLOAD(3, k0 + 28);
    }
    WMMA4(0);
    WMMA4(1);
    WMMA4(2);
    WMMA4(3);
#undef PLOAD

    const float bi0 = bias[n0 +      r];
    const float bi1 = bias[n0 + 16 + r];
    const float bi2 = bias[n0 + 32 + r];
    const float bi3 = bias[n0 + 48 + r];
#pragma unroll
    for (int i = 0; i < 8; ++i) {
        const int m = m0 + i + 8 * half;          // C/D layout: VGPR i -> row i (+8 for hi lanes)
        float* yr = Y + (size_t)m * N + n0 + r;
        yr[0]  = acc0[i] + bi0;
        yr[16] = acc1[i] + bi1;
        yr[32] = acc2[i] + bi2;
        yr[48] = acc3[i] + bi3;
    }
}

// ---------------------------------------------------------------------------
// scores[t,b,s] = sum_d swish(pi[t,b,d] + pc[b,s,d]) * w_one[d] + b_one
// Block handles 4 t-values x all 128 s for one b. pi rows + w_one in LDS.
// ---------------------------------------------------------------------------
__global__ __launch_bounds__(256) void swish_score_kernel(
    const float* __restrict__ pi,    // [T*B, D]
    const float* __restrict__ pc,    // [B*S, D]
    const float* __restrict__ w_one, // [D]
    const float* __restrict__ b_one, // scalar
    float* __restrict__ scores)      // [T*B, S]
{
    __shared__ float pi_s[4][D];
    __shared__ float w_s[D];

    const int b  = blockIdx.x & (B - 1);
    const int t0 = (blockIdx.x >> 4) * 4;

    for (int idx = threadIdx.x; idx < 4 * D; idx += 256) {
        const int tt = idx >> 10, d = idx & (D - 1);
        pi_s[tt][d] = pi[((size_t)(t0 + tt) * B + b) * D + d];
    }
    for (int d = threadIdx.x; d < D; d += 256)
        w_s[d] = w_one[d];
    __syncthreads();

    const int wave = threadIdx.x >> 5;
    const int lane = threadIdx.x & 31;
    const float bone = b_one[0];

    for (int s = wave; s < S; s += 8) {
        const float* pcr = pc + ((size_t)b * S + s) * D;
        float a0 = 0.f, a1 = 0.f, a2 = 0.f, a3 = 0.f;
        for (int d = lane; d < D; d += 32) {
            const float pcv = pcr[d];
            const float wv  = w_s[d];
            float x;
            x = pi_s[0][d] + pcv; a0 += wv * x * fast_sigmoid(x);
            x = pi_s[1][d] + pcv; a1 += wv * x * fast_sigmoid(x);
            x = pi_s[2][d] + pcv; a2 += wv * x * fast_sigmoid(x);
            x = pi_s[3][d] + pcv; a3 += wv * x * fast_sigmoid(x);
        }
#pragma unroll
        for (int off = 16; off; off >>= 1) {
            a0 += __shfl_xor(a0, off, 32);
            a1 += __shfl_xor(a1, off, 32);
            a2 += __shfl_xor(a2, off, 32);
            a3 += __shfl_xor(a3, off, 32);
        }
        if (lane == 0) {
            scores[((size_t)(t0 + 0) * B + b) * S + s] = a0 + bone;
            scores[((size_t)(t0 + 1) * B + b) * S + s] = a1 + bone;
            scores[((size_t)(t0 + 2) * B + b) * S + s] = a2 + bone;
            scores[((size_t)(t0 + 3) * B + b) * S + s] = a3 + bone;
        }
    }
}

// ---------------------------------------------------------------------------
// Row softmax over S=128; one wave per (t,b) row, 4 elements per lane.
// ---------------------------------------------------------------------------
__global__ __launch_bounds__(32) void softmax_kernel(
    const float* __restrict__ scores, float* __restrict__ attn)
{
    const int row  = blockIdx.x;          // t*B + b
    const int lane = threadIdx.x;
    const float* sr = scores + (size_t)row * S;

    float v[4];
    float mx = -3.402823466e38f;
#pragma unroll
    for (int i = 0; i < 4; ++i) { v[i] = sr[lane + 32 * i]; mx = fmaxf(mx, v[i]); }
#pragma unroll
    for (int off = 16; off; off >>= 1) mx = fmaxf(mx, __shfl_xor(mx, off, 32));
    float sum = 0.f;
#pragma unroll
    for (int i = 0; i < 4; ++i) { v[i] = __expf(v[i] - mx); sum += v[i]; }
#pragma unroll
    for (int off = 16; off; off >>= 1) sum += __shfl_xor(sum, off, 32);
    const float rs = __builtin_amdgcn_rcpf(sum);
    float* ar = attn + (size_t)row * S;
#pragma unroll
    for (int i = 0; i < 4; ++i) ar[lane + 32 * i] = v[i] * rs;
}

// ---------------------------------------------------------------------------
// out[t,b,e] = sum_s attn[t,b,s] * context[b,s,e]   (fp32 WMMA, per-b GEMM)
// M=T=32, K=S=128, N=D=1024 per batch b. Wave tile 16x64, 4-deep pipeline.
// ---------------------------------------------------------------------------
__global__ __launch_bounds__(256) void attn_ctx_kernel(
    const float* __restrict__ attn,     // [T,B,S]
    const float* __restrict__ context,  // [B,S,D]
    float* __restrict__ out)            // [T,B,D]
{
    const int wid  = (blockIdx.x * blockDim.x + threadIdx.x) >> 5;
    const int lane = threadIdx.x & 31;
    const int half = lane >> 4;
    const int r    = lane & 15;
    const int b    = wid >> 5;            // 32 wave tiles per batch (2 M x 16 N)
    const int rem  = wid & 31;
    const int m0   = (rem >> 4) * 16;
    const int n0   = (rem & 15) * 64;
    const int ko   = 2 * half;

    const float* Arow = attn + ((size_t)(m0 + r) * B + b) * S;
    const float* Cb   = context + (size_t)b * S * D;

#define ALOAD(sl, kk)                                                                              \
    a##sl = LD2(Arow + (kk) + ko);                                                                 \
    {                                                                                              \
        const float* p0 = Cb + (size_t)((kk) + ko) * D + n0 + r;                                   \
        const float* p1 = p0 + D;                                                                  \
        s0##sl = (v2f){p0[0],  p1[0]};                                                             \
        s1##sl = (v2f){p0[16], p1[16]};                                                            \
        s2##sl = (v2f){p0[32], p1[32]};                                                            \
        s3##sl = (v2f){p0[48], p1[48]};                                                            \
    }

    v8f acc0 = {}, acc1 = {}, acc2 = {}, acc3 = {};
    v2f a0, s00, s10, s20, s30;
    v2f a1, s01, s11, s21, s31;
    v2f a2, s02, s12, s22, s32;
    v2f a3, s03, s13, s23, s33;

    ALOAD(0, 0);
    ALOAD(1, 4);
    ALOAD(2, 8);
    ALOAD(3, 12);
    for (int k0 = 0; k0 < S - 16; k0 += 16) {
        WMMA4(0); ALOAD(0, k0 + 16);
        WMMA4(1); ALOAD(1, k0 + 20);
        WMMA4(2); ALOAD(2, k0 + 24);
        WMMA4(3); ALOAD(3, k0 + 28);
    }
    WMMA4(0);
    WMMA4(1);
    WMMA4(2);
    WMMA4(3);
#undef ALOAD

#pragma unroll
    for (int i = 0; i < 8; ++i) {
        const int m = m0 + i + 8 * half;
        float* yr = out + ((size_t)m * B + b) * D + n0 + r;
        yr[0]  = acc0[i];
        yr[16] = acc1[i];
        yr[32] = acc2[i];
        yr[48] = acc3[i];
    }
}

// ---------------------------------------------------------------------------
extern "C" void kernel_launch(void* const* d_in, const int* in_sizes, int n_in,
                              void* d_out, int out_size, void* d_ws, size_t ws_size,
                              hipStream_t stream)
{
    const float* inputs  = (const float*)d_in[0];   // [T,B,D]
    const float* context = (const float*)d_in[1];   // [B,S,D]
    const float* W_in    = (const float*)d_in[2];   // [D,D]
    const float* b_in    = (const float*)d_in[3];   // [D]
    const float* W_ctx   = (const float*)d_in[4];   // [D,D]
    const float* b_ctx   = (const float*)d_in[5];   // [D]
    const float* w_one   = (const float*)d_in[6];   // [D]
    const float* b_one   = (const float*)d_in[7];   // scalar

    float* out      = (float*)d_out;                 // attn_context [T,B,D]
    float* attn_out = out + (size_t)T * B * D;       // attn [T,B,S]

    float* pi     = (float*)d_ws;                    // [T*B, D]  2 MB
    float* pc     = pi + (size_t)T * B * D;          // [B*S, D]  8 MB
    float* scores = pc + (size_t)B * S * D;          // [T*B, S]  256 KB

    // 1) pi = inputs @ W_in^T + b_in : M=512 -> 512 waves -> 64 blocks
    proj_gemm_kernel<<<64, 256, 0, stream>>>(inputs, W_in, b_in, pi, T * B);
    // 2) pc = context @ W_ctx^T + b_ctx : M=2048 -> 2048 waves -> 256 blocks
    proj_gemm_kernel<<<256, 256, 0, stream>>>(context, W_ctx, b_ctx, pc, B * S);
    // 3) fused add + swish + dot(w_one): 128 blocks = (T/4) * B
    swish_score_kernel<<<(T / 4) * B, 256, 0, stream>>>(pi, pc, w_one, b_one, scores);
    // 4) softmax rows -> attn (second output), one wave per row
    softmax_kernel<<<T * B, 32, 0, stream>>>(scores, attn_out);
    // 5) attn @ context : 512 waves -> 64 blocks
    attn_ctx_kernel<<<64, 256, 0, stream>>>(attn_out, context, out);
}